// PoolingSpanExtractor_48576080118507
// MI455X (gfx1250) — compile-verified
//
#include <hip/hip_runtime.h>
#include <stdint.h>

// Problem constants (fixed by the reference file)
#define BQ   8
#define SQ   2048
#define NQ   4096
#define DQ   256
#define MAXW 16

#define BDIM   128           // 4 wave32s per block
#define WAVES  (BDIM / 32)

typedef uint32_t u32;
typedef __attribute__((ext_vector_type(4))) u32   u32x4;
typedef __attribute__((ext_vector_type(8))) int   i32x8;
typedef __attribute__((ext_vector_type(4))) int   i32x4;
typedef __attribute__((ext_vector_type(4))) float f32x4;

#if defined(__gfx1250__) && __has_builtin(__builtin_amdgcn_tensor_load_to_lds)
#define USE_TDM 1
#else
#define USE_TDM 0
#endif

__global__ __launch_bounds__(BDIM)
void PoolingSpanExtractor_48576080118507_kernel(const float* __restrict__ seq,
                                                const int* __restrict__ spans,
                                                float* __restrict__ out,
                                                int n_spans)
{
    // 4 waves * 16 rows * 256 f32 = 64 KB static LDS (WGP has 320 KB)
    __shared__ __align__(16) float smem[WAVES * MAXW * DQ];

    const int lane = threadIdx.x & 31;
    const int wave = threadIdx.x >> 5;
    const int span = blockIdx.x * WAVES + wave;      // one span per wave
    if (span >= n_spans) return;

    // Wave-uniform span descriptor: contiguous rows start..end of seq[b]
    const int start = __builtin_amdgcn_readfirstlane(spans[2 * span + 0]);
    const int end   = __builtin_amdgcn_readfirstlane(spans[2 * span + 1]);
    const int cnt   = end - start + 1;               // 1..16 (width in [0,16))

    const int b = span / NQ;                          // batch index
    const uint64_t gaddr =
        (uint64_t)(uintptr_t)(seq + ((uint64_t)b * SQ + (uint64_t)start) * DQ);

    float* wbuf = &smem[wave * (MAXW * DQ)];

#if USE_TDM
    // ---- Tensor Data Mover: DMA the (cnt x 256) f32 tile into this wave's LDS ----
    // D# group 0 (CDNA5 ISA)
    const u32 lds_addr = (u32)(uintptr_t)wbuf;        // flat LDS addr: low 32 bits = LDS byte offset
    u32x4 g0;
    g0.x = 1u;                                        // count=1, is_restore=0, gather_mode=0
    g0.y = lds_addr;                                  // lds_addr (bytes)
    g0.z = (u32)gaddr;                                // global_addr[31:0]
    g0.w = ((u32)(gaddr >> 32) & 0x01FFFFFFu)         // global_addr[56:32]
         | (2u << 30);                                // type=2 ("image")

    // D# group 1 (§8.4)
    i32x8 g1;
    g1[0] = (int)(2u << 16);                          // workgroup_mask=0, data_size=2 (4 bytes)
    g1[1] = (int)((u32)DQ << 16);                     // tensor_dim0[15:0]=256 in bits 63:48
    g1[2] = (int)((u32)SQ << 16);                     // tensor_dim0 hi=0 | tensor_dim1[15:0]=2048
    g1[3] = (int)((u32)DQ << 16);                     // tensor_dim1 hi=0 | tile_dim0=256
    g1[4] = cnt;                                      // tile_dim1 = rows (1..16), tile_dim2=0
    g1[5] = DQ;                                       // tensor_dim0_stride = 256 elements
    g1[6] = 0;                                        // stride hi / tensor_dim1_stride lo
    g1[7] = 0;

    i32x4 g2 = {0, 0, 0, 0};                          // 2-D tile: groups 2/3 unused
    i32x4 g3 = {0, 0, 0, 0};
    i32x8 g4 = {0, 0, 0, 0, 0, 0, 0, 0};              // extra group (clang-23 6-arg form)

    __builtin_amdgcn_tensor_load_to_lds(g0, g1, g2, g3, g4, /*cpol=*/0);
    __builtin_amdgcn_s_wait_tensorcnt(0);             // wave-local: only this wave reads wbuf
#else
    // Fallback: per-lane vector loads into LDS (same layout)
    for (int r = 0; r < cnt; ++r) {
        const f32x4* src = (const f32x4*)(seq + ((uint64_t)b * SQ + (uint64_t)(start + r)) * DQ);
        f32x4* dst = (f32x4*)(wbuf + r * DQ);
        dst[lane * 2 + 0] = src[lane * 2 + 0];
        dst[lane * 2 + 1] = src[lane * 2 + 1];
    }
    __syncthreads();
#endif

    // ---- Reduce: each lane owns 8 columns (two b128 LDS loads per row) ----
    const f32x4* buf = (const f32x4*)wbuf;
    f32x4 a0 = {0.f, 0.f, 0.f, 0.f};
    f32x4 a1 = {0.f, 0.f, 0.f, 0.f};
    #pragma unroll 1
    for (int r = 0; r < cnt; ++r) {
        a0 += buf[r * (DQ / 4) + lane * 2 + 0];
        a1 += buf[r * (DQ / 4) + lane * 2 + 1];
    }
    const float scale = 1.0f / (float)cnt;
    a0 *= scale;
    a1 *= scale;

    f32x4* op = (f32x4*)(out + (uint64_t)span * DQ);
    op[lane * 2 + 0] = a0;
    op[lane * 2 + 1] = a1;
}

extern "C" void kernel_launch(void* const* d_in, const int* in_sizes, int n_in,
                              void* d_out, int out_size, void* d_ws, size_t ws_size,
                              hipStream_t stream) {
    const float* seq   = (const float*)d_in[0];   // (B,S,D) f32
    const int*   spans = (const int*)d_in[1];     // (B,N,2) i32
    float*       out   = (float*)d_out;           // (B,N,D) f32

    const int n_spans = in_sizes[1] / 2;          // B*N = 32768
    const int blocks  = (n_spans + WAVES - 1) / WAVES;

    PoolingSpanExtractor_48576080118507_kernel<<<blocks, BDIM, 0, stream>>>(
        seq, spans, out, n_spans);
}